// UpDownLayer_25933012533790
// MI455X (gfx1250) — compile-verified
//
#include <hip/hip_runtime.h>

// Gather-GEMM: out[n, :] = concat_{j=0..8}( x[adjc[n,j], :] ) @ W
//   x:    [196608, 64] f32   (50 MB -> resident in 192 MB L2, gathers are L2 hits)
//   adjc: [196608, 9]  i32
//   W:    [576, 64]    f32
//   out:  [196608, 64] f32
//
// GEMM [N x 576] @ [576 x 64] with gathered A rows, V_WMMA_F32_16X16X4_F32
// (exact f32 path, wave32). Double-buffered LDS pipeline, one barrier/stage.
// Staging uses GLOBAL_LOAD_ASYNC_TO_LDS_B128 (ASYNCcnt) when the toolchain
// exposes it, else global->VGPR->ds_store_b128.

#define N_CELLS 196608
#define NH      9
#define F_IN    64
#define F_OUT   64

typedef float v2f __attribute__((ext_vector_type(2)));
typedef float v8f __attribute__((ext_vector_type(8)));

// LDS row strides (floats), bank-conflict-free for the WMMA fragment reads:
//  A: stride 68 -> lanes m=0..15 hit banks {4m,4m+1} (+2 for hi half): all 64.
//  B: stride 80 -> hi=1 half-wave offset by 32 banks: disjoint windows.
#define A_STRIDE 68
#define W_STRIDE 80
#define A_ELEMS  (32 * A_STRIDE)   // 8704 B per buffer
#define W_ELEMS  (64 * W_STRIDE)   // 20480 B per buffer

#if defined(__has_builtin)
#  if __has_builtin(__builtin_amdgcn_global_load_async_to_lds_b128)
#    define HAVE_ASYNC_LDS 1
#  endif
#endif
#ifndef HAVE_ASYNC_LDS
#  define HAVE_ASYNC_LDS 0
#endif

#if HAVE_ASYNC_LDS
#  if __has_builtin(__builtin_amdgcn_s_wait_asynccnt)
#    define WAIT_ASYNC() __builtin_amdgcn_s_wait_asynccnt(0)
#  else
#    define WAIT_ASYNC() asm volatile("s_wait_asynccnt 0x0" ::: "memory")
#  endif
// Signature learned from the round-3 diagnostic: param 1 is a NON-const
// AS(1) pointer to 'int __attribute__((vector_size(16)))'; 4 args total.
typedef int b128_t __attribute__((vector_size(16)));
#  define GLB_CAST(p) ((__attribute__((address_space(1))) b128_t*)(p))
#  define LDS_CAST(p) ((__attribute__((address_space(3))) b128_t*)(p))
#endif

__global__ __launch_bounds__(256)
void updown_gather_gemm(const float* __restrict__ x,
                        const int*   __restrict__ adjc,
                        const float* __restrict__ W,
                        float*       __restrict__ out)
{
    __shared__ float As[2][A_ELEMS];
    __shared__ float Ws[2][W_ELEMS];

    const int t    = threadIdx.x;
    const int lane = t & 31;
    const int wave = t >> 5;        // 8 waves
    const int lo   = lane & 15;
    const int hi   = lane >> 4;
    const int mt   = wave >> 2;     // M-tile (0..1)
    const int nt   = wave & 3;      // N-tile (0..3)
    const int cellBase = blockIdx.x * 32;

    // Staging geometry: thread t owns A float4-slots {t, t+256} and W slots
    // {t + 256*i : i<4}.  Slot idx -> row idx>>4, float4 idx&15.
    const int c0 = t >> 4;            // 0..15
    const int c1 = (t + 256) >> 4;    // 16..31
    const int c4 = t & 15;
    const int w0 = t >> 4;
    const int w1 = (t + 256) >> 4;
    const int w2 = (t + 512) >> 4;
    const int w3 = (t + 768) >> 4;

    // Preload all adjacency indices this thread stages (read-once -> NT loads);
    // removes the adjc->x dependent-load chain from the pipelined stages.
    int r0[NH], r1[NH];
#pragma unroll
    for (int j = 0; j < NH; ++j) {
        r0[j] = __builtin_nontemporal_load(&adjc[(cellBase + c0) * NH + j]);
        r1[j] = __builtin_nontemporal_load(&adjc[(cellBase + c1) * NH + j]);
    }

    v8f acc = {0.f, 0.f, 0.f, 0.f, 0.f, 0.f, 0.f, 0.f};

#if HAVE_ASYNC_LDS
    // ---- async staging: 6x GLOBAL_LOAD_ASYNC_TO_LDS_B128 per thread/stage ----
#define ISSUE_STAGE(J, B)                                                          \
    do {                                                                           \
        __builtin_amdgcn_global_load_async_to_lds_b128(                            \
            GLB_CAST(x + (long)r0[(J)] * F_IN + c4 * 4),                           \
            LDS_CAST(&As[(B)][c0 * A_STRIDE + c4 * 4]), 0, 0);                     \
        __builtin_amdgcn_global_load_async_to_lds_b128(                            \
            GLB_CAST(x + (long)r1[(J)] * F_IN + c4 * 4),                           \
            LDS_CAST(&As[(B)][c1 * A_STRIDE + c4 * 4]), 0, 0);                     \
        __builtin_amdgcn_global_load_async_to_lds_b128(                            \
            GLB_CAST(W + ((J) * F_IN + w0) * F_OUT + c4 * 4),                      \
            LDS_CAST(&Ws[(B)][w0 * W_STRIDE + c4 * 4]), 0, 0);                     \
        __builtin_amdgcn_global_load_async_to_lds_b128(                            \
            GLB_CAST(W + ((J) * F_IN + w1) * F_OUT + c4 * 4),                      \
            LDS_CAST(&Ws[(B)][w1 * W_STRIDE + c4 * 4]), 0, 0);                     \
        __builtin_amdgcn_global_load_async_to_lds_b128(                            \
            GLB_CAST(W + ((J) * F_IN + w2) * F_OUT + c4 * 4),                      \
            LDS_CAST(&Ws[(B)][w2 * W_STRIDE + c4 * 4]), 0, 0);                     \
        __builtin_amdgcn_global_load_async_to_lds_b128(                            \
            GLB_CAST(W + ((J) * F_IN + w3) * F_OUT + c4 * 4),                      \
            LDS_CAST(&Ws[(B)][w3 * W_STRIDE + c4 * 4]), 0, 0);                     \
    } while (0)

    ISSUE_STAGE(0, 0);
    WAIT_ASYNC();
    __syncthreads();
#else
    // ---- register staging fallback: global->VGPR, ds_store_b128 after compute ----
    float4 ax0, ax1, wx0, wx1, wx2, wx3;
#define LOAD_STAGE(J)                                                              \
    do {                                                                           \
        ax0 = ((const float4*)(x + (long)r0[(J)] * F_IN))[c4];                     \
        ax1 = ((const float4*)(x + (long)r1[(J)] * F_IN))[c4];                     \
        wx0 = ((const float4*)(W + ((J) * F_IN + w0) * F_OUT))[c4];                \
        wx1 = ((const float4*)(W + ((J) * F_IN + w1) * F_OUT))[c4];                \
        wx2 = ((const float4*)(W + ((J) * F_IN + w2) * F_OUT))[c4];                \
        wx3 = ((const float4*)(W + ((J) * F_IN + w3) * F_OUT))[c4];                \
    } while (0)
#define STORE_STAGE(B)                                                             \
    do {                                                                           \
        *(float4*)&As[(B)][c0 * A_STRIDE + c4 * 4] = ax0;                          \
        *(float4*)&As[(B)][c1 * A_STRIDE + c4 * 4] = ax1;                          \
        *(float4*)&Ws[(B)][w0 * W_STRIDE + c4 * 4] = wx0;                          \
        *(float4*)&Ws[(B)][w1 * W_STRIDE + c4 * 4] = wx1;                          \
        *(float4*)&Ws[(B)][w2 * W_STRIDE + c4 * 4] = wx2;                          \
        *(float4*)&Ws[(B)][w3 * W_STRIDE + c4 * 4] = wx3;                          \
    } while (0)

    LOAD_STAGE(0);
    STORE_STAGE(0);
    __syncthreads();
#endif

#pragma unroll
    for (int j = 0; j < NH; ++j) {
        const int cur = j & 1;

#if HAVE_ASYNC_LDS
        // issue next stage's async copies into the other buffer; they fill LDS
        // while the WMMAs below run. Safe: last readers of that buffer finished
        // before the end-of-(j-1) barrier.
        if (j + 1 < NH) ISSUE_STAGE(j + 1, (j + 1) & 1);
#else
        if (j + 1 < NH) LOAD_STAGE(j + 1);
#endif

        // 16 x V_WMMA_F32_16X16X4_F32 over this K=64 block.
        // A 16x4 f32: lane l -> m=l&15; VGPR0=A[m][2*hi], VGPR1=A[m][2*hi+1]
        // B 4x16 f32: lane l -> n=l&15; VGPR0=B[2*hi][n], VGPR1=B[2*hi+1][n]
        const float* arow  = &As[cur][(mt * 16 + lo) * A_STRIDE];
        const float* wbase = &Ws[cur][nt * 16 + lo];
#pragma unroll
        for (int kk = 0; kk < 16; ++kk) {
            const int k = kk * 4;
            v2f a, b;
            a.x = arow[k + 2 * hi];
            a.y = arow[k + 2 * hi + 1];
            b.x = wbase[(k + 2 * hi    ) * W_STRIDE];
            b.y = wbase[(k + 2 * hi + 1) * W_STRIDE];
            acc = __builtin_amdgcn_wmma_f32_16x16x4_f32(
                false, a, false, b, (short)0, acc, false, false);
        }

        if (j + 1 < NH) {
#if HAVE_ASYNC_LDS
            WAIT_ASYNC();               // stage j+1 copies landed in LDS
#else
            STORE_STAGE((j + 1) & 1);
#endif
            __syncthreads();
        }
    }

    // store 16x16 f32 D tile (VGPR v: lanes 0-15 -> M=v; 16-31 -> M=v+8), NT
    // stores so the 50 MB output stream doesn't evict x from L2.
    float* o = out + (long)(cellBase + mt * 16) * F_OUT + nt * 16 + lo;
#pragma unroll
    for (int v = 0; v < 8; ++v) {
        __builtin_nontemporal_store(acc[v], &o[(v + 8 * hi) * F_OUT]);
    }
}

extern "C" void kernel_launch(void* const* d_in, const int* in_sizes, int n_in,
                              void* d_out, int out_size, void* d_ws, size_t ws_size,
                              hipStream_t stream) {
    (void)in_sizes; (void)n_in; (void)d_ws; (void)ws_size; (void)out_size;
    const float* x    = (const float*)d_in[0];
    const int*   adjc = (const int*)d_in[1];
    const float* W    = (const float*)d_in[2];
    float*       out  = (float*)d_out;

    const int blocks = N_CELLS / 32;   // 6144 workgroups, 256 thr (8 waves)
    updown_gather_gemm<<<blocks, 256, 0, stream>>>(x, adjc, W, out);
}